// GNN_77120432767032
// MI455X (gfx1250) — compile-verified
//
#include <hip/hip_runtime.h>
#include <math.h>

typedef __attribute__((ext_vector_type(2))) float v2f;
typedef __attribute__((ext_vector_type(8))) float v8f;

#define N_NODES 100000
#define N_EDGES 3200000
#define D_IN    128
#define D_H     16
#define D_OUT   2

// ---------------- degree / normalization ----------------

__global__ void k_init_deg(float* __restrict__ deg) {
    int n = blockIdx.x * blockDim.x + threadIdx.x;
    if (n < N_NODES) deg[n] = 1.0f;   // self-loop contributes 1 to degree
}

__global__ void k_deg_accum(const long long* __restrict__ ei, float* __restrict__ deg) {
    int e = blockIdx.x * blockDim.x + threadIdx.x;
    if (e < N_EDGES) {
        int dst = (int)ei[N_EDGES + e];           // col = target
        atomicAdd(&deg[dst], 1.0f);
    }
}

__global__ void k_dis(float* __restrict__ deg) {
    int n = blockIdx.x * blockDim.x + threadIdx.x;
    if (n < N_NODES) {
        float d = deg[n];
        deg[n] = (d > 0.0f) ? rsqrtf(d) : 0.0f;   // in place: deg -> deg^{-1/2}
    }
}

// ---------------- GEMM1: h = x @ W1  (WMMA f32 16x16x4) ----------------
// One wave per 16-row tile. A: 16x128 rows of x, B: 128x16 = W1, C/D: 16x16 f32.

__global__ void __launch_bounds__(256) k_gemm1(const float* __restrict__ x,
                                               const float* __restrict__ W1,
                                               float* __restrict__ h) {
    const int wave = threadIdx.x >> 5;
    const int lane = threadIdx.x & 31;
    const int tile = blockIdx.x * (blockDim.x >> 5) + wave;
    if (tile * 16 >= N_NODES) return;             // wave-uniform: EXEC stays all-1s

    const int half = lane >> 4;                   // 0: lanes 0-15, 1: lanes 16-31
    const int l16  = lane & 15;
    const int m    = tile * 16 + l16;             // A row handled by this lane
    const int koff = half * 2;                    // K sub-offset per ISA f32 A/B layout

    v8f acc = {};
#pragma unroll
    for (int kb = 0; kb < D_IN; kb += 4) {
        const int k = kb + koff;
        v2f a, b;
        a.x = x[m * D_IN + k];                    // A[m][k],   A[m][k+1]
        a.y = x[m * D_IN + k + 1];
        b.x = W1[(k)     * D_H + l16];            // B[k][n],   B[k+1][n]
        b.y = W1[(k + 1) * D_H + l16];
        acc = __builtin_amdgcn_wmma_f32_16x16x4_f32(
                  false, a, false, b, (short)0, acc, false, false);
    }
    // D layout: VGPR r holds row (r + 8*half), col l16
#pragma unroll
    for (int r = 0; r < 8; ++r) {
        int mm = tile * 16 + r + 8 * half;
        h[mm * D_H + l16] = acc[r];
    }
}

// ---------------- layer-1 aggregation ----------------

__global__ void k_self16(const float* __restrict__ h, const float* __restrict__ dis,
                         float* __restrict__ agg) {
    int idx = blockIdx.x * blockDim.x + threadIdx.x;   // over N_NODES*16
    if (idx < N_NODES * D_H) {
        int n = idx >> 4;
        float s = dis[n];
        agg[idx] = h[idx] * s * s;                 // self-loop: norm = dis[n]^2
    }
}

__global__ void k_edge16(const long long* __restrict__ ei,
                         const float* __restrict__ h, const float* __restrict__ dis,
                         float* __restrict__ agg) {
    int idx = blockIdx.x * blockDim.x + threadIdx.x;   // over N_EDGES*16
    if (idx < N_EDGES * D_H) {
        int e = idx >> 4, c = idx & 15;
        int src = (int)ei[e];
        int dst = (int)ei[N_EDGES + e];
        float v = h[src * D_H + c] * dis[src] * dis[dst];
        atomicAdd(&agg[dst * D_H + c], v);
    }
}

// ---------------- GEMM2: h2 = relu(agg1 + b1) @ W2  (WMMA, fused relu+bias) ----

__global__ void __launch_bounds__(256) k_gemm2(const float* __restrict__ agg1,
                                               const float* __restrict__ b1,
                                               const float* __restrict__ W2,
                                               float* __restrict__ h2) {
    const int wave = threadIdx.x >> 5;
    const int lane = threadIdx.x & 31;
    const int tile = blockIdx.x * (blockDim.x >> 5) + wave;
    if (tile * 16 >= N_NODES) return;

    const int half = lane >> 4;
    const int l16  = lane & 15;
    const int m    = tile * 16 + l16;
    const int koff = half * 2;

    v8f acc = {};
#pragma unroll
    for (int kb = 0; kb < D_H; kb += 4) {
        const int k = kb + koff;
        v2f a, b;
        a.x = fmaxf(agg1[m * D_H + k]     + b1[k],     0.0f);   // fused relu+bias
        a.y = fmaxf(agg1[m * D_H + k + 1] + b1[k + 1], 0.0f);
        b.x = (l16 < D_OUT) ? W2[(k)     * D_OUT + l16] : 0.0f; // W2 padded 16x2->16x16
        b.y = (l16 < D_OUT) ? W2[(k + 1) * D_OUT + l16] : 0.0f;
        acc = __builtin_amdgcn_wmma_f32_16x16x4_f32(
                  false, a, false, b, (short)0, acc, false, false);
    }
    if (l16 < D_OUT) {
#pragma unroll
        for (int r = 0; r < 8; ++r) {
            int mm = tile * 16 + r + 8 * half;
            h2[mm * D_OUT + l16] = acc[r];
        }
    }
}

// ---------------- layer-2 aggregation + log-softmax ----------------

__global__ void k_self2(const float* __restrict__ h2, const float* __restrict__ dis,
                        float* __restrict__ agg) {
    int idx = blockIdx.x * blockDim.x + threadIdx.x;   // over N_NODES*2
    if (idx < N_NODES * D_OUT) {
        int n = idx >> 1;
        float s = dis[n];
        agg[idx] = h2[idx] * s * s;
    }
}

__global__ void k_edge2(const long long* __restrict__ ei,
                        const float* __restrict__ h2, const float* __restrict__ dis,
                        float* __restrict__ agg) {
    int idx = blockIdx.x * blockDim.x + threadIdx.x;   // over N_EDGES*2
    if (idx < N_EDGES * D_OUT) {
        int e = idx >> 1, c = idx & 1;
        int src = (int)ei[e];
        int dst = (int)ei[N_EDGES + e];
        float v = h2[src * D_OUT + c] * dis[src] * dis[dst];
        atomicAdd(&agg[dst * D_OUT + c], v);
    }
}

__global__ void k_final(const float* __restrict__ agg2, const float* __restrict__ b2,
                        float* __restrict__ out) {
    int n = blockIdx.x * blockDim.x + threadIdx.x;
    if (n < N_NODES) {
        float z0 = agg2[n * 2 + 0] + b2[0];
        float z1 = agg2[n * 2 + 1] + b2[1];
        float mx = fmaxf(z0, z1);
        float lse = mx + logf(expf(z0 - mx) + expf(z1 - mx));
        out[n * 2 + 0] = z0 - lse;
        out[n * 2 + 1] = z1 - lse;
    }
}

// ---------------- driver ----------------

extern "C" void kernel_launch(void* const* d_in, const int* in_sizes, int n_in,
                              void* d_out, int out_size, void* d_ws, size_t ws_size,
                              hipStream_t stream) {
    const float*      x  = (const float*)d_in[0];
    const long long*  ei = (const long long*)d_in[1];   // [2, E] int64
    const float*      W1 = (const float*)d_in[2];
    const float*      b1 = (const float*)d_in[3];
    const float*      W2 = (const float*)d_in[4];
    const float*      b2 = (const float*)d_in[5];
    float* out = (float*)d_out;

    float* ws   = (float*)d_ws;
    float* dis  = ws;                        //  N  (deg, then deg^-1/2 in place)
    float* h    = dis  + N_NODES;            // 16N
    float* agg1 = h    + N_NODES * D_H;      // 16N
    float* h2   = agg1 + N_NODES * D_H;      //  2N
    float* agg2 = h2   + N_NODES * D_OUT;    //  2N   -> total 37N floats (~14.8 MB)

    const int B = 256;
    const int ntiles = N_NODES / 16;         // 6250, exact

    k_init_deg <<<(N_NODES + B - 1) / B, B, 0, stream>>>(dis);
    k_deg_accum<<<(N_EDGES + B - 1) / B, B, 0, stream>>>(ei, dis);
    k_dis      <<<(N_NODES + B - 1) / B, B, 0, stream>>>(dis);

    k_gemm1 <<<(ntiles + 7) / 8, 256, 0, stream>>>(x, W1, h);
    k_self16<<<(N_NODES * D_H + B - 1) / B, B, 0, stream>>>(h, dis, agg1);
    k_edge16<<<(N_EDGES * D_H + B - 1) / B, B, 0, stream>>>(ei, h, dis, agg1);

    k_gemm2<<<(ntiles + 7) / 8, 256, 0, stream>>>(agg1, b1, W2, h2);
    k_self2<<<(N_NODES * D_OUT + B - 1) / B, B, 0, stream>>>(h2, dis, agg2);
    k_edge2<<<(N_EDGES * D_OUT + B - 1) / B, B, 0, stream>>>(ei, h2, dis, agg2);

    k_final<<<(N_NODES + B - 1) / B, B, 0, stream>>>(agg2, b2, out);
}